// SinkhornDistance_62491774157231
// MI455X (gfx1250) — compile-verified
//
#include <hip/hip_runtime.h>

// ---------------------------------------------------------------------------
// Sinkhorn distance, MI455X (gfx1250), wave32.
//   B=16, n_s=n_t=512, d=64, fp32.
// Phase 1: cost/K/K^T via V_WMMA_F32_16X16X4_F32 (fp32 matrix pipe -> exact
//          reference precision; exp(-10*cost) is exponentially sensitive).
// Phase 2: 100 iterations of u = mu/(K v), v = nu/(K^T u). K & K^T are
//          L2-resident (32 MB << 192 MB L2); one 1024-thread WG per batch,
//          wave-per-row coalesced b128 reads, LDS-held u/v.
// Phase 3: plan = u*K*v streamed to HBM; distance via fixed-order reductions
//          (deterministic, no float atomics).
// ---------------------------------------------------------------------------

typedef float v2f __attribute__((ext_vector_type(2)));
typedef float v4f __attribute__((ext_vector_type(4)));
typedef float v8f __attribute__((ext_vector_type(8)));

static constexpr int    BATCH   = 16;
static constexpr int    N       = 512;   // n_s == n_t
static constexpr int    DIM     = 64;
static constexpr float  REG_INV = 10.0f; // 1 / 0.1
static constexpr int    ITERS   = 100;

// Workspace layout (float offsets)
static constexpr size_t K_OFF  = 0;                               // 16 MB
static constexpr size_t KT_OFF = (size_t)BATCH * N * N;           // 16 MB
static constexpr size_t C_OFF  = 2 * (size_t)BATCH * N * N;       // 16 MB
static constexpr size_t SN_OFF = 3 * (size_t)BATCH * N * N;       // row norms of S
static constexpr size_t TN_OFF = SN_OFF + (size_t)BATCH * N;      // row norms of T
static constexpr size_t U_OFF  = TN_OFF + (size_t)BATCH * N;
static constexpr size_t V_OFF  = U_OFF  + (size_t)BATCH * N;
static constexpr size_t RS_OFF = V_OFF  + (size_t)BATCH * N;      // per-row dist partials

// ---------------------------------------------------------------------------
// Kernel 1: squared row norms of source and target. One wave per row.
// ---------------------------------------------------------------------------
__global__ __launch_bounds__(256) void rownorm_kernel(const float* __restrict__ src,
                                                      const float* __restrict__ tgt,
                                                      float* __restrict__ ws) {
    const int wave = (blockIdx.x * 256 + threadIdx.x) >> 5;   // 16384 waves
    const int lane = threadIdx.x & 31;
    const int isT  = wave >> 13;                              // first 8192 -> S
    const int row  = wave & 8191;                             // flat b*512+r
    const float* p = (isT ? tgt : src) + (size_t)row * DIM + lane * 2;
    v2f x = *(const v2f*)p;
    float s = x.x * x.x + x.y * x.y;
#pragma unroll
    for (int off = 16; off > 0; off >>= 1) s += __shfl_xor(s, off, 32);
    if (lane == 0) ws[(isT ? TN_OFF : SN_OFF) + row] = s;
}

// ---------------------------------------------------------------------------
// Kernel 2: G = S*T^T via fp32 WMMA, then cost = sn+tn-2G, K = exp(-cost/reg),
// writing cost, K and K^T. One wave per 16x16 output tile.
//
// V_WMMA_F32_16X16X4_F32 fragment layout (ISA 7.12.2):
//   A (16x4): lane = M (0-15 twice); half-wave selects K={0,1} vs K={2,3};
//             VGPR0=K even, VGPR1=K odd  -> per-lane float2 at (row, k0+2*hf).
//   B (4x16): symmetric with N in place of M.
//   C/D (16x16): VGPR r -> M = r + 8*hf, N = lane&15.
// ---------------------------------------------------------------------------
__global__ __launch_bounds__(256) void costK_kernel(const float* __restrict__ src,
                                                    const float* __restrict__ tgt,
                                                    float* __restrict__ ws) {
    const int wave = (blockIdx.x * 256 + threadIdx.x) >> 5;   // 16384 tile-waves
    const int lane = threadIdx.x & 31;
    const int b    = wave >> 10;                              // 1024 tiles / batch
    const int t    = wave & 1023;
    const int i0   = (t >> 5) * 16;
    const int j0   = (t & 31) * 16;
    const int hf   = lane >> 4;                               // half-wave selector
    const int rc   = lane & 15;                               // A row / B col in tile

    const float* Sb = src + ((size_t)b * N + i0 + rc) * DIM + 2 * hf;
    const float* Tb = tgt + ((size_t)b * N + j0 + rc) * DIM + 2 * hf;

    v8f c = {0.f, 0.f, 0.f, 0.f, 0.f, 0.f, 0.f, 0.f};
#pragma unroll
    for (int k0 = 0; k0 < DIM; k0 += 4) {
        v2f a  = *(const v2f*)(Sb + k0);
        v2f bb = *(const v2f*)(Tb + k0);
        // 8 args: (neg_a, A, neg_b, B, c_mod, C, reuse_a, reuse_b)
        c = __builtin_amdgcn_wmma_f32_16x16x4_f32(false, a, false, bb,
                                                  (short)0, c, false, false);
    }

    const float tnv = ws[TN_OFF + (size_t)b * N + j0 + rc];
    float* Kp  = ws + K_OFF;
    float* KTp = ws + KT_OFF;
    float* Cp  = ws + C_OFF;
#pragma unroll
    for (int r = 0; r < 8; ++r) {
        const int   m    = r + 8 * hf;
        const int   n    = rc;
        const float snv  = ws[SN_OFF + (size_t)b * N + i0 + m];
        const float cost = snv + tnv - 2.0f * c[r];
        const float kv   = expf(-cost * REG_INV);
        const size_t idx  = ((size_t)b * N + (i0 + m)) * N + (j0 + n);
        const size_t idxT = ((size_t)b * N + (j0 + n)) * N + (i0 + m);
        Cp[idx]   = cost;
        Kp[idx]   = kv;
        KTp[idxT] = kv;   // one-time scattered store; phase 2 reads stay coalesced
    }
}

// ---------------------------------------------------------------------------
// Kernel 3: 100 Sinkhorn iterations. One workgroup (32 wave32s) per batch.
// u, v in LDS; each wave owns 16 rows; lanes stride columns with b128 loads.
// ---------------------------------------------------------------------------
__global__ __launch_bounds__(1024) void sinkhorn_kernel(float* __restrict__ ws) {
    __shared__ __align__(16) float u_lds[N];
    __shared__ __align__(16) float v_lds[N];

    const int b    = blockIdx.x;
    const int tid  = threadIdx.x;
    const int wave = tid >> 5;
    const int lane = tid & 31;
    const float mu = 1.0f / N;
    const float nu = 1.0f / N;

    if (tid < N) v_lds[tid] = 1.0f;   // v0 = ones (u0 overwritten in step 1)
    __syncthreads();

    const float* Kb  = ws + K_OFF  + (size_t)b * N * N;
    const float* KTb = ws + KT_OFF + (size_t)b * N * N;

    for (int it = 0; it < ITERS; ++it) {
        // u = mu / (K v)
#pragma unroll 4
        for (int rr = 0; rr < 16; ++rr) {
            const int row = wave * 16 + rr;
            const float* Kr = Kb + (size_t)row * N;
            float s = 0.f;
#pragma unroll
            for (int jt = 0; jt < 4; ++jt) {
                const int j0 = jt * 128 + lane * 4;
                v4f kk = *(const v4f*)(Kr + j0);
                v4f vv = *(const v4f*)(v_lds + j0);
                s += kk.x * vv.x + kk.y * vv.y + kk.z * vv.z + kk.w * vv.w;
            }
#pragma unroll
            for (int off = 16; off > 0; off >>= 1) s += __shfl_xor(s, off, 32);
            if (lane == 0) u_lds[row] = mu / s;
        }
        __syncthreads();

        // v = nu / (K^T u)
#pragma unroll 4
        for (int rr = 0; rr < 16; ++rr) {
            const int col = wave * 16 + rr;
            const float* Kc = KTb + (size_t)col * N;
            float s = 0.f;
#pragma unroll
            for (int jt = 0; jt < 4; ++jt) {
                const int i0 = jt * 128 + lane * 4;
                v4f kk = *(const v4f*)(Kc + i0);
                v4f uu = *(const v4f*)(u_lds + i0);
                s += kk.x * uu.x + kk.y * uu.y + kk.z * uu.z + kk.w * uu.w;
            }
#pragma unroll
            for (int off = 16; off > 0; off >>= 1) s += __shfl_xor(s, off, 32);
            if (lane == 0) v_lds[col] = nu / s;
        }
        __syncthreads();
    }

    if (tid < N) ws[U_OFF + (size_t)b * N + tid] = u_lds[tid];
    if (tid < N) ws[V_OFF + (size_t)b * N + tid] = v_lds[tid];
}

// ---------------------------------------------------------------------------
// Kernel 4: plan = u * K * v (streamed to d_out), per-row distance partials.
// One wave per row; deterministic in-wave reduction order.
// ---------------------------------------------------------------------------
__global__ __launch_bounds__(256) void plan_kernel(float* __restrict__ ws,
                                                   float* __restrict__ out) {
    const int wave = (blockIdx.x * 256 + threadIdx.x) >> 5;   // 8192 row-waves
    const int lane = threadIdx.x & 31;
    const int b    = wave >> 9;
    const int row  = wave & 511;

    const float ub = ws[U_OFF + (size_t)b * N + row];
    const float* Kr = ws + K_OFF + ((size_t)b * N + row) * N;
    const float* Cr = ws + C_OFF + ((size_t)b * N + row) * N;
    const float* Vb = ws + V_OFF + (size_t)b * N;
    float* Pr = out + BATCH + ((size_t)b * N + row) * N;      // plan after distance[16]

    float s = 0.f;
#pragma unroll
    for (int jt = 0; jt < 4; ++jt) {
        const int j0 = jt * 128 + lane * 4;
        v4f kk = *(const v4f*)(Kr + j0);
        v4f cc = *(const v4f*)(Cr + j0);
        v4f vv = *(const v4f*)(Vb + j0);
        v4f p;
        p.x = ub * kk.x * vv.x;
        p.y = ub * kk.y * vv.y;
        p.z = ub * kk.z * vv.z;
        p.w = ub * kk.w * vv.w;
        *(v4f*)(Pr + j0) = p;
        s += p.x * cc.x + p.y * cc.y + p.z * cc.z + p.w * cc.w;
    }
#pragma unroll
    for (int off = 16; off > 0; off >>= 1) s += __shfl_xor(s, off, 32);
    if (lane == 0) ws[RS_OFF + (size_t)b * N + row] = s;
}

// ---------------------------------------------------------------------------
// Kernel 5: distance[b] = fixed-order tree reduction of 512 row partials.
// ---------------------------------------------------------------------------
__global__ __launch_bounds__(256) void dist_kernel(const float* __restrict__ ws,
                                                   float* __restrict__ out) {
    __shared__ float red[256];
    const int b = blockIdx.x;
    const int t = threadIdx.x;
    red[t] = ws[RS_OFF + (size_t)b * N + t] + ws[RS_OFF + (size_t)b * N + t + 256];
    __syncthreads();
#pragma unroll
    for (int off = 128; off > 0; off >>= 1) {
        if (t < off) red[t] += red[t + off];
        __syncthreads();
    }
    if (t == 0) out[b] = red[0];
}

// ---------------------------------------------------------------------------
extern "C" void kernel_launch(void* const* d_in, const int* in_sizes, int n_in,
                              void* d_out, int out_size, void* d_ws, size_t ws_size,
                              hipStream_t stream) {
    (void)in_sizes; (void)n_in; (void)out_size; (void)ws_size;
    const float* src = (const float*)d_in[0];   // [16,512,64]
    const float* tgt = (const float*)d_in[1];   // [16,512,64]
    float* out = (float*)d_out;                 // [16] dist ++ [16,512,512] plan
    float* ws  = (float*)d_ws;                  // ~48.2 MB used

    rownorm_kernel <<<2048, 256, 0, stream>>>(src, tgt, ws);
    costK_kernel   <<<2048, 256, 0, stream>>>(src, tgt, ws);
    sinkhorn_kernel<<<BATCH, 1024, 0, stream>>>(ws);
    plan_kernel    <<<1024, 256, 0, stream>>>(ws, out);
    dist_kernel    <<<BATCH, 256, 0, stream>>>(ws, out);
}